// GraphGaussianPolicyNetworkBuilder_17781164605529
// MI455X (gfx1250) — compile-verified
//
#include <hip/hip_runtime.h>

// GraphGaussianPolicyNetwork forward on gfx1250 (MI455X), fp32 WMMA path.
// One 16-batch tile per 4-wave workgroup; activations in a shared 96KB LDS
// slab pair -> 3 workgroups (12 waves) per WGP. Two independent WMMA
// accumulator chains for ILP in stages 1-5 (stage 6 uses one).

typedef __attribute__((ext_vector_type(2))) float v2f;
typedef __attribute__((ext_vector_type(8))) float v8f;

#define BATCH   65536
#define NN      48                      // graph nodes
#define NA      16                      // action nodes
#define SLAB    (NN * 16 * 16)          // floats per H buffer (12288 = 48KB)
#define NWAVES  4

static __device__ __forceinline__ v8f wmma4(v2f a, v2f b, v8f c) {
  // D = A(16x4 f32) * B(4x16 f32) + C(16x16 f32)
  return __builtin_amdgcn_wmma_f32_16x16x4_f32(
      /*neg_a=*/false, a, /*neg_b=*/false, b,
      /*c_mod=*/(short)0, c, /*reuse_a=*/false, /*reuse_b=*/false);
}

// In-place feature transform on this wave's 12 nodes, 2 nodes per step:
// buf[j][h][b] = sum_d W[d][h] * buf[j][d][b]
static __device__ __forceinline__ void feat_part(float* buf, const float* __restrict__ gw,
                                                 int wave, int ln, int half) {
  v2f at[4];
#pragma unroll
  for (int kk = 0; kk < 4; ++kk) {            // A = W^T : A[m=h][k=d]
    int d0 = 4 * kk + 2 * half;
    at[kk].x = gw[(d0 + 0) * 16 + ln];
    at[kk].y = gw[(d0 + 1) * 16 + ln];
  }
  for (int jj = 0; jj < 12; jj += 2) {
    float* p0 = buf + (wave * 12 + jj) * 256;
    float* p1 = p0 + 256;
    v2f b0v[4], b1v[4];
#pragma unroll
    for (int kk = 0; kk < 4; ++kk) {          // B[k=d][n=b] from LDS
      int d0 = 4 * kk + 2 * half;
      b0v[kk].x = p0[(d0 + 0) * 16 + ln];  b0v[kk].y = p0[(d0 + 1) * 16 + ln];
      b1v[kk].x = p1[(d0 + 0) * 16 + ln];  b1v[kk].y = p1[(d0 + 1) * 16 + ln];
    }
    v8f c0 = {}, c1 = {};
#pragma unroll
    for (int kk = 0; kk < 4; ++kk) {          // two independent chains
      c0 = wmma4(at[kk], b0v[kk], c0);
      c1 = wmma4(at[kk], b1v[kk], c1);
    }
#pragma unroll
    for (int r = 0; r < 8; ++r) {             // C row m = r + 8*half
      p0[(r + 8 * half) * 16 + ln] = c0[r];
      p1[(r + 8 * half) * 16 + ln] = c1[r];
    }
  }
}

// dst[i][h][b] = relu( sum_j clip(A[i][j]) * src[j][h][b] + gb[h] )
// This wave handles h = wave*4 .. wave*4+3 (2 at a time), all it tiles.
static __device__ __forceinline__ void node_mix_part(const float* __restrict__ src,
                                                     float* __restrict__ dst,
                                                     const float* __restrict__ Amat,
                                                     const float* __restrict__ gb,
                                                     int itMax, int wave, int ln, int half) {
  for (int it = 0; it < itMax; ++it) {
    v2f av[12];
#pragma unroll
    for (int kk = 0; kk < 12; ++kk) {         // A = clipped adjacency tile, K = j
      int i  = it * 16 + ln;
      int j0 = 4 * kk + 2 * half;
      av[kk].x = fminf(fmaxf(Amat[i * NN + j0],     -1.f), 1.f);
      av[kk].y = fminf(fmaxf(Amat[i * NN + j0 + 1], -1.f), 1.f);
    }
#pragma unroll
    for (int hh = 0; hh < 4; hh += 2) {
      int h = wave * 4 + hh;
      float bias0 = gb[h], bias1 = gb[h + 1];
      v8f c0, c1;
#pragma unroll
      for (int r = 0; r < 8; ++r) { c0[r] = bias0; c1[r] = bias1; }
#pragma unroll
      for (int kk = 0; kk < 12; ++kk) {       // B[k=j][n=b] at fixed h / h+1
        int j0 = 4 * kk + 2 * half;
        v2f b0v, b1v;
        b0v.x = src[(j0 + 0) * 256 + h * 16 + ln];
        b0v.y = src[(j0 + 1) * 256 + h * 16 + ln];
        b1v.x = src[(j0 + 0) * 256 + (h + 1) * 16 + ln];
        b1v.y = src[(j0 + 1) * 256 + (h + 1) * 16 + ln];
        c0 = wmma4(av[kk], b0v, c0);
        c1 = wmma4(av[kk], b1v, c1);
      }
#pragma unroll
      for (int r = 0; r < 8; ++r) {
        int row = it * 16 + r + 8 * half;
        dst[row * 256 + h * 16 + ln]       = fmaxf(c0[r], 0.f);
        dst[row * 256 + (h + 1) * 16 + ln] = fmaxf(c1[r], 0.f);
      }
    }
  }
}

// In-place per-node PFC on this wave's 4 nodes (2 at a time):
// buf[n][o][b] = relu( sum_i pw[n][o][i]*buf[n][i][b] + pb[n][o] )
static __device__ __forceinline__ void pfc_part(float* buf, const float* __restrict__ pw,
                                                const float* __restrict__ pb,
                                                int wave, int ln, int half) {
#pragma unroll
  for (int nn2 = 0; nn2 < 4; nn2 += 2) {
    int n0 = wave * 4 + nn2, n1 = n0 + 1;
    float* p0 = buf + n0 * 256;
    float* p1 = buf + n1 * 256;
    v2f a0v[4], a1v[4], b0v[4], b1v[4];
#pragma unroll
    for (int kk = 0; kk < 4; ++kk) {
      int i0 = 4 * kk + 2 * half;
      a0v[kk].x = pw[(n0 * 16 + ln) * 16 + i0];      // A[m=o][k=i]
      a0v[kk].y = pw[(n0 * 16 + ln) * 16 + i0 + 1];
      a1v[kk].x = pw[(n1 * 16 + ln) * 16 + i0];
      a1v[kk].y = pw[(n1 * 16 + ln) * 16 + i0 + 1];
      b0v[kk].x = p0[(i0 + 0) * 16 + ln];            // B[k=i][n=b]
      b0v[kk].y = p0[(i0 + 1) * 16 + ln];
      b1v[kk].x = p1[(i0 + 0) * 16 + ln];
      b1v[kk].y = p1[(i0 + 1) * 16 + ln];
    }
    v8f c0, c1;
#pragma unroll
    for (int r = 0; r < 8; ++r) {
      c0[r] = pb[n0 * 16 + r + 8 * half];
      c1[r] = pb[n1 * 16 + r + 8 * half];
    }
#pragma unroll
    for (int kk = 0; kk < 4; ++kk) {
      c0 = wmma4(a0v[kk], b0v[kk], c0);
      c1 = wmma4(a1v[kk], b1v[kk], c1);
    }
#pragma unroll
    for (int r = 0; r < 8; ++r) {
      p0[(r + 8 * half) * 16 + ln] = fmaxf(c0[r], 0.f);
      p1[(r + 8 * half) * 16 + ln] = fmaxf(c1[r], 0.f);
    }
  }
}

__global__ __launch_bounds__(NWAVES * 32)
void policy_fwd_kernel(const float* __restrict__ obs,     const float* __restrict__ embed_w,
                       const float* __restrict__ embed_b, const float* __restrict__ Amat,
                       const float* __restrict__ gcn_w,   const float* __restrict__ gcn_b,
                       const float* __restrict__ pfc_w,   const float* __restrict__ pfc_b,
                       const float* __restrict__ mean_w,  const float* __restrict__ logstd_w,
                       float* __restrict__ out) {
  extern __shared__ float lds[];
  const int lane = threadIdx.x & 31;
  const int wave = threadIdx.x >> 5;     // 0..3, owns nodes [12w,12w+12), h/n [4w,4w+4)
  const int ln   = lane & 15;            // batch column / tile row index
  const int half = lane >> 4;            // K-pair select (A/B), row-half select (C)
  const int b0   = blockIdx.x * 16;      // one 16-batch tile per workgroup

  float* buf0 = lds;
  float* buf1 = lds + SLAB;

  // ---- Stage 1: embed + GCN-1 feature transform on this wave's 12 nodes
  for (int jj = 0; jj < 12; jj += 4) {
    int jb = wave * 12 + jj;             // 4-aligned
    float4 ob4 = *(const float4*)(obs + (b0 + ln) * NN + jb);
    float obv[4] = {ob4.x, ob4.y, ob4.z, ob4.w};
#pragma unroll
    for (int u = 0; u < 4; ++u) {
      int j = jb + u, t = j >> 4;
#pragma unroll
      for (int k = 0; k < 8; ++k) {
        int d = 2 * k + half;
        buf0[j * 256 + d * 16 + ln] = obv[u] * embed_w[t * 16 + d] + embed_b[t * 16 + d];
      }
    }
  }
  feat_part(buf0, gcn_w, wave, ln, half);            // same-wave nodes: no barrier yet
  __syncthreads();

  // ---- Stage 2: GCN-1 node mix (all 48 output nodes), buf0 -> buf1
  node_mix_part(buf0, buf1, Amat, gcn_b, 3, wave, ln, half);
  __syncthreads();

  // ---- Stage 3: GCN-2 feature transform, in-place on buf1
  feat_part(buf1, gcn_w + 256, wave, ln, half);
  __syncthreads();

  // ---- Stage 4: GCN-2 node mix, only nodes 0..15 are consumed downstream
  node_mix_part(buf1, buf0, Amat, gcn_b + 16, 1, wave, ln, half);
  __syncthreads();

  // ---- Stage 5: PFC layer 1, in-place on this wave's 4 nodes
  pfc_part(buf0, pfc_w, pfc_b, wave, ln, half);
  // same node ownership in stage 6 -> no barrier needed

  // ---- Stage 6: PFC layer 2 fused with mean/log_std heads
  const float* pw2 = pfc_w + 4096;      // layer stride 16*16*16
  const float* pb2 = pfc_b + 256;       // layer stride 16*16
#pragma unroll
  for (int nn2 = 0; nn2 < 4; ++nn2) {
    int n = wave * 4 + nn2;
    float* p = buf0 + n * 256;
    v2f av[4], bv[4];
#pragma unroll
    for (int kk = 0; kk < 4; ++kk) {
      int i0 = 4 * kk + 2 * half;
      av[kk].x = pw2[(n * 16 + ln) * 16 + i0];
      av[kk].y = pw2[(n * 16 + ln) * 16 + i0 + 1];
      bv[kk].x = p[(i0 + 0) * 16 + ln];
      bv[kk].y = p[(i0 + 1) * 16 + ln];
    }
    v8f c;
#pragma unroll
    for (int r = 0; r < 8; ++r) c[r] = pb2[n * 16 + r + 8 * half];
#pragma unroll
    for (int kk = 0; kk < 4; ++kk) c = wmma4(av[kk], bv[kk], c);

    float pm = 0.f, pl = 0.f;
#pragma unroll
    for (int r = 0; r < 8; ++r) {
      float x = fmaxf(c[r], 0.f);
      int o = r + 8 * half;
      pm += x * mean_w[n * 16 + o];
      pl += x * logstd_w[n * 16 + o];
    }
    pm += __shfl_xor(pm, 16, 32);       // combine row halves (wave32)
    pl += __shfl_xor(pl, 16, 32);
    pl = fminf(fmaxf(pl, -5.f), 2.f);
    if (half == 0) out[(b0 + ln) * NA + n] = pm;                // mean
    else           out[BATCH * NA + (b0 + ln) * NA + n] = pl;   // log_std
  }
}

extern "C" void kernel_launch(void* const* d_in, const int* in_sizes, int n_in,
                              void* d_out, int out_size, void* d_ws, size_t ws_size,
                              hipStream_t stream) {
  (void)in_sizes; (void)n_in; (void)out_size; (void)d_ws; (void)ws_size;
  const float* obs      = (const float*)d_in[0];
  const float* embed_w  = (const float*)d_in[1];
  const float* embed_b  = (const float*)d_in[2];
  const float* Amat     = (const float*)d_in[3];
  const float* gcn_w    = (const float*)d_in[4];
  const float* gcn_b    = (const float*)d_in[5];
  const float* pfc_w    = (const float*)d_in[6];
  const float* pfc_b    = (const float*)d_in[7];
  const float* mean_w   = (const float*)d_in[8];
  const float* logstd_w = (const float*)d_in[9];
  float* out = (float*)d_out;

  dim3 grid(BATCH / 16);                 // 4096 workgroups, one 16-batch tile each
  dim3 block(NWAVES * 32);               // 128 threads = 4 waves
  size_t shmem = (size_t)2 * SLAB * sizeof(float);   // 96 KB -> 3 blocks / WGP
  policy_fwd_kernel<<<grid, block, shmem, stream>>>(
      obs, embed_w, embed_b, Amat, gcn_w, gcn_b, pfc_w, pfc_b, mean_w, logstd_w, out);
}